// LocalGlobalBlock_69303592288306
// MI455X (gfx1250) — compile-verified
//
#include <hip/hip_runtime.h>
#include <math.h>

// ---------------- problem constants (from reference) ----------------
#define SEQL   512
#define SAMP   4
#define HID    128
#define DH     64
#define KCONV  3
#define SEG    128      // SEQ / SAMP
#define BS     64
#define CS     32
#define NROWS  (BS * CS)        // 2048
#define NLAY   10

typedef __attribute__((ext_vector_type(2))) float v2f;
typedef __attribute__((ext_vector_type(8))) float v8f;

__device__ __forceinline__ float gelu_exact(float v) {
    return 0.5f * v * (1.0f + erff(v * 0.70710678118654752f));
}

// =====================================================================
// Kernel 1: fused MaskMLP on m = f_weight * x, written to d_out.
//   grid = NROWS/16 blocks, block = 128 threads = 4 waves (wave w <-> s=w)
//
//   Stage0: stage m-tile [16 x 512] into LDS, transposed per sub-sample:
//           ms[s][j][row]  (A-fragment reads are then consecutive-word,
//           bank-conflict-free; the K+2 half-wave lands +32 words away)
//   Stage1: mid[bc,s,h] = gelu( sum_j m[bc,j*4+s]*wes1[h,j*4+s] + b1 )
//           via V_WMMA_F32_16X16X4_F32, result staged as midT[s][h][row]
//   Stage2: out[bc,g*4+s] = sum_h mid*wes2[(s*128+g), s*128+h] + b2
//                           + res[bc,g]  (res computed inline from ms)
// =====================================================================
__global__ __launch_bounds__(128) void mask_mlp_wmma(
    const float* __restrict__ x,    const float* __restrict__ fw,
    const float* __restrict__ wes1, const float* __restrict__ bias1,
    const float* __restrict__ wes2, const float* __restrict__ bias2,
    const float* __restrict__ wes3, const float* __restrict__ bias3,
    float* __restrict__ out)
{
    __shared__ float ms[SAMP][SEG][16];    // 32 KB : m, transposed per s
    __shared__ float midT[SAMP][HID][16];  // 32 KB : gelu(stage1), transposed

    const int tid  = threadIdx.x;
    const int s    = tid >> 5;           // wave id == sub-sample index
    const int lane = tid & 31;
    const int bc0  = blockIdx.x * 16;

    // warm L2 for the broadcast weight panels
    __builtin_prefetch(&wes1[0], 0, 3);
    __builtin_prefetch(&wes2[(long)s * SEG * SEQL], 0, 3);

    // ---- Stage 0: coalesced staging of m = fw*x (computed once) ----
    for (int i = tid; i < 16 * SEQL; i += 128) {
        const int row = i >> 9;          // 0..15
        const int col = i & 511;         // 0..511
        const long g  = (long)(bc0 + row) * SEQL + col;
        ms[col & 3][col >> 2][row] = fw[g] * x[g];
    }
    __syncthreads();

    const int hl  = lane & 15;           // A row / B col within tile
    const int kb  = (lane >> 4) * 2;     // K sub-offset per ISA layout
    const int rof = (lane >> 4) * 8;     // C/D row offset

    // ---- Stage 1: mid = gelu(m_s @ W1_s^T + b1) ----
    for (int nt = 0; nt < 8; ++nt) {
        v8f acc = {};
        const int h0 = nt * 16 + hl;
        for (int k0 = 0; k0 < SEG; k0 += 4) {
            const int j0 = k0 + kb;
            v2f a, b;
            a.x = ms[s][j0][hl];                 // conflict-free LDS reads
            a.y = ms[s][j0 + 1][hl];
            const long ib = (long)h0 * SEQL + (long)j0 * 4 + s;
            b.x = wes1[ib];
            b.y = wes1[ib + 4];
            acc = __builtin_amdgcn_wmma_f32_16x16x4_f32(
                      false, a, false, b, (short)0, acc, false, false);
        }
        const float b1 = bias1[s * HID + h0];
        #pragma unroll
        for (int r = 0; r < 8; ++r) {
            const int row = r + rof;
            midT[s][h0][row] = gelu_exact(acc[r] + b1);
        }
    }
    __syncthreads();

    // ---- Stage 2: out = mid @ W2diag_s^T + b2 + res ----
    for (int nt = 0; nt < 8; ++nt) {
        v8f acc = {};
        const int g0 = nt * 16 + hl;
        const long brow = (long)(s * SEG + g0) * SEQL + s * HID;
        for (int k0 = 0; k0 < HID; k0 += 4) {
            const int kk = k0 + kb;
            v2f a, b;
            a.x = midT[s][kk][hl];               // conflict-free LDS reads
            a.y = midT[s][kk + 1][hl];
            b.x = wes2[brow + kk];
            b.y = wes2[brow + kk + 1];
            acc = __builtin_amdgcn_wmma_f32_16x16x4_f32(
                      false, a, false, b, (short)0, acc, false, false);
        }
        // per-lane constants for this output column g0
        const float b2 = bias2[s * SEG + g0];
        float w3v0 = wes3[g0 * 516 + 0];         // wes3[g][g*4+sp]
        float w3v1 = wes3[g0 * 516 + 1];
        float w3v2 = wes3[g0 * 516 + 2];
        float w3v3 = wes3[g0 * 516 + 3];
        const float b3 = bias3[g0];
        #pragma unroll
        for (int r = 0; r < 8; ++r) {
            const int row = r + rof;
            float res = b3;
            res = fmaf(ms[0][g0][row], w3v0, res);
            res = fmaf(ms[1][g0][row], w3v1, res);
            res = fmaf(ms[2][g0][row], w3v2, res);
            res = fmaf(ms[3][g0][row], w3v3, res);
            out[(long)(bc0 + row) * SEQL + g0 * 4 + s] = acc[r] + b2 + res;
        }
    }
}

// =====================================================================
// Kernel 2: DecomposedBlock pyramid convs; accumulates into d_out.
//   grid = NROWS blocks (one signal row each), block = 256 threads.
//   Segments: [0,256) depth4 (layers 6..9), [256,384) depth3 (3..5),
//             [384,448) depth2 (1..2),      [448,512) depth1 (0).
// =====================================================================
__global__ __launch_bounds__(256) void decomposed_conv(
    const float* __restrict__ x,
    const float* __restrict__ c1w, const float* __restrict__ c1b,
    const float* __restrict__ c2w, const float* __restrict__ c2b,
    float* __restrict__ out)
{
    __shared__ float h[SEQL];
    __shared__ float w1s[NLAY * DH * KCONV];   // 1920
    __shared__ float b1s[NLAY * DH];           // 640
    __shared__ float w2s[NLAY * DH];           // 640
    __shared__ float b2s[NLAY];

    const int  tid = threadIdx.x;
    const long row = blockIdx.x;

    for (int i = tid; i < NLAY * DH * KCONV; i += 256) w1s[i] = c1w[i];
    for (int i = tid; i < NLAY * DH; i += 256) { b1s[i] = c1b[i]; w2s[i] = c2w[i]; }
    if (tid < NLAY) b2s[tid] = c2b[tid];
    for (int p = tid; p < SEQL; p += 256) h[p] = x[row * SEQL + p];
    __syncthreads();

    for (int it = 0; it < 4; ++it) {
        float nv[2];
        bool  act[2];
        #pragma unroll
        for (int q = 0; q < 2; ++q) {
            const int p = tid + q * 256;
            int loff, start, len, depth;
            if (p < 256)      { loff = 6; start = 0;   len = 256; depth = 4; }
            else if (p < 384) { loff = 3; start = 256; len = 128; depth = 3; }
            else if (p < 448) { loff = 1; start = 384; len = 64;  depth = 2; }
            else              { loff = 0; start = 448; len = 64;  depth = 1; }
            act[q] = (it < depth);
            nv[q] = 0.0f;
            if (act[q]) {
                const int li = loff + it;
                const int l  = p - start;
                const float hm = (l > 0)       ? h[p - 1] : 0.0f;
                const float h0 = h[p];
                const float hp = (l < len - 1) ? h[p + 1] : 0.0f;
                const float* w1 = &w1s[li * DH * KCONV];
                const float* b1 = &b1s[li * DH];
                const float* w2 = &w2s[li * DH];
                float acc = b2s[li];
                #pragma unroll 4
                for (int c = 0; c < DH; ++c) {
                    float z = fmaf(w1[c * 3 + 0], hm,
                              fmaf(w1[c * 3 + 1], h0,
                              fmaf(w1[c * 3 + 2], hp, b1[c])));
                    acc = fmaf(w2[c], gelu_exact(z), acc);
                }
                nv[q] = acc + h0;   // residual
            }
        }
        __syncthreads();
        #pragma unroll
        for (int q = 0; q < 2; ++q) {
            const int p = tid + q * 256;
            if (act[q]) h[p] = nv[q];
        }
        __syncthreads();
    }

    for (int p = tid; p < SEQL; p += 256) {
        const long o = row * SEQL + p;
        out[o] = out[o] + h[p];     // add l_out on top of g_out from kernel 1
    }
}

// =====================================================================
extern "C" void kernel_launch(void* const* d_in, const int* in_sizes, int n_in,
                              void* d_out, int out_size, void* d_ws, size_t ws_size,
                              hipStream_t stream) {
    (void)in_sizes; (void)n_in; (void)out_size; (void)d_ws; (void)ws_size;
    const float* x     = (const float*)d_in[0];
    const float* fw    = (const float*)d_in[1];
    const float* c1w   = (const float*)d_in[2];
    const float* c1b   = (const float*)d_in[3];
    const float* c2w   = (const float*)d_in[4];
    const float* c2b   = (const float*)d_in[5];
    const float* wes1  = (const float*)d_in[6];
    const float* bias1 = (const float*)d_in[7];
    const float* wes2  = (const float*)d_in[8];
    const float* bias2 = (const float*)d_in[9];
    const float* wes3  = (const float*)d_in[10];
    const float* bias3 = (const float*)d_in[11];
    float* out = (float*)d_out;

    mask_mlp_wmma<<<NROWS / 16, 128, 0, stream>>>(
        x, fw, wes1, bias1, wes2, bias2, wes3, bias3, out);
    decomposed_conv<<<NROWS, 256, 0, stream>>>(
        x, c1w, c1b, c2w, c2b, out);
}